// GPFPlus_2869038154348
// MI455X (gfx1250) — compile-verified
//
#include <hip/hip_runtime.h>
#include <hip/hip_bf16.h>

// Problem constants (match reference)
#define NN 64
#define BB 256
#define DD 1792
#define KIN 35
#define PP 10
#define PD (PP * DD)      // 17920
#define COLS (PD + DD)    // 19712 : [prompt_bank cols | ptok cols]

typedef float v2f __attribute__((ext_vector_type(2)));
typedef float v8f __attribute__((ext_vector_type(8)));
typedef int   v4i __attribute__((ext_vector_type(4)));

#define AS1 __attribute__((address_space(1)))
#define AS3 __attribute__((address_space(3)))

#if __has_builtin(__builtin_amdgcn_global_load_async_to_lds_b128)
#define HAVE_ASYNC_LDS 1
#else
#define HAVE_ASYNC_LDS 0
#endif

__device__ __forceinline__ void wait_asynccnt0() {
#if __has_builtin(__builtin_amdgcn_s_wait_asynccnt)
    __builtin_amdgcn_s_wait_asynccnt(0);
#else
    asm volatile("s_wait_asynccnt 0x0" ::: "memory");
#endif
}

// ---------------------------------------------------------------------------
// Kernel A: fused GEMM  out[b, j] = sum_k task_id[b,k] * W[j,k] + bias[j]
// cols j<PD use (W_tok, b_tok) -> prompt bank, j>=PD use (W_ptok, b_ptok).
// One 16x16 output tile per wave via V_WMMA_F32_16X16X4_F32, K padded 35->36.
// ---------------------------------------------------------------------------
__global__ __launch_bounds__(256) void gpf_bank_wmma(
    const float* __restrict__ task_id,
    const float* __restrict__ W_tok,  const float* __restrict__ b_tok,
    const float* __restrict__ W_ptok, const float* __restrict__ b_ptok,
    float* __restrict__ ws_bank,      // (BB, PD)
    float* __restrict__ ws_ptok)      // (BB, DD)
{
    const int CT   = COLS / 16;                       // 1232 col tiles
    const int wave = blockIdx.x * 8 + (threadIdx.x >> 5);
    const int lane = threadIdx.x & 31;
    const int tj = wave % CT, tb = wave / CT;
    const int m0 = tb * 16, j0 = tj * 16;
    const int half = lane >> 4, mn = lane & 15;

    const int m  = m0 + mn;          // A row held by this lane
    const int jn = j0 + mn;          // B/C column held by this lane
    const bool isBank = (j0 < PD);   // whole tile in one region (PD % 16 == 0)

    const float* __restrict__ arow = task_id + (size_t)m * KIN;
    const float* __restrict__ wrow = isBank ? (W_tok  + (size_t)jn * KIN)
                                            : (W_ptok + (size_t)(jn - PD) * KIN);

    v8f acc = {};
    #pragma unroll
    for (int k0 = 0; k0 < 36; k0 += 4) {
        const int ka = k0 + half * 2;   // this lane's K pair (A and B symmetric)
        v2f a, bm;
        a.x  = (ka     < KIN) ? arow[ka]     : 0.0f;
        a.y  = (ka + 1 < KIN) ? arow[ka + 1] : 0.0f;
        bm.x = (ka     < KIN) ? wrow[ka]     : 0.0f;
        bm.y = (ka + 1 < KIN) ? wrow[ka + 1] : 0.0f;
        acc = __builtin_amdgcn_wmma_f32_16x16x4_f32(
            /*neg_a=*/false, a, /*neg_b=*/false, bm,
            /*c_mod=*/(short)0, acc, /*reuse_a=*/false, /*reuse_b=*/false);
    }

    const float bias = isBank ? b_tok[jn] : b_ptok[jn - PD];
    #pragma unroll
    for (int r = 0; r < 8; ++r) {
        const int mr = m0 + half * 8 + r;     // C/D layout: M = 8*(lane/16) + r
        const float v = acc[r] + bias;
        if (isBank) ws_bank[(size_t)mr * PD + jn] = v;
        else        ws_ptok[(size_t)mr * DD + (jn - PD)] = v;
    }
}

// ---------------------------------------------------------------------------
// Kernel B: fused LayerNorm + sigmoid gate + prompt einsum + residual.
// One block per b; bank[b] (70KB) + ptok[b] (7KB) staged into LDS via
// GLOBAL_LOAD_ASYNC_TO_LDS_B128 (ASYNCcnt path); W_net slice lives in
// registers across the 64-row n loop. X read once, out written once.
// ---------------------------------------------------------------------------
__global__ __launch_bounds__(256) void gpf_fused_main(
    const float* __restrict__ X,
    const float* __restrict__ gamma, const float* __restrict__ beta,
    const float* __restrict__ W_net, const float* __restrict__ b_net,
    const float* __restrict__ ws_bank, const float* __restrict__ ws_ptok,
    float* __restrict__ out)
{
    __shared__ float s_bank[PP * DD];     // 71680 B
    __shared__ float s_ptok[DD];          //  7168 B
    __shared__ float s_red[2][8];         // per-wave {sum, sumsq}
    __shared__ float s_dot[8][PP];        // per-wave dot partials
    __shared__ float s_w[PP];             // broadcast gate weights
    __shared__ float s_bnet[PP];

    const int b    = blockIdx.x;
    const int tid  = threadIdx.x;
    const int lane = tid & 31;
    const int wv   = tid >> 5;

#if HAVE_ASYNC_LDS
    {
        const float* gsrc = ws_bank + (size_t)b * PD;
        for (int i = tid * 4; i < PP * DD; i += 256 * 4)
            __builtin_amdgcn_global_load_async_to_lds_b128(
                (v4i*)(gsrc + i), (AS3 v4i*)(&s_bank[i]), 0, 0);
        const float* psrc = ws_ptok + (size_t)b * DD;
        for (int i = tid * 4; i < DD; i += 256 * 4)
            __builtin_amdgcn_global_load_async_to_lds_b128(
                (v4i*)(psrc + i), (AS3 v4i*)(&s_ptok[i]), 0, 0);
    }
#else
    for (int i = tid; i < PP * DD; i += 256) s_bank[i] = ws_bank[(size_t)b * PD + i];
    for (int i = tid; i < DD; i += 256)      s_ptok[i] = ws_ptok[(size_t)b * DD + i];
#endif
    if (tid < PP) s_bnet[tid] = b_net[tid];

    // Per-thread d-slice: d = tid + 256*i, i in [0,7)  (DD = 7*256, coalesced)
    float g[7], be[7], wn[PP][7];
    #pragma unroll
    for (int i = 0; i < 7; ++i) {
        const int d = tid + 256 * i;
        g[i] = gamma[d]; be[i] = beta[d];
    }
    #pragma unroll
    for (int p = 0; p < PP; ++p)
        #pragma unroll
        for (int i = 0; i < 7; ++i)
            wn[p][i] = W_net[p * DD + tid + 256 * i];

#if HAVE_ASYNC_LDS
    wait_asynccnt0();      // this wave's async LDS writes landed
#endif
    __syncthreads();       // all waves' staging visible block-wide

    const float invD = 1.0f / (float)DD;

    for (int n = 0; n < NN; ++n) {
        const float* __restrict__ xrow = X   + ((size_t)n * BB + b) * (size_t)DD;
        float*       __restrict__ orow = out + ((size_t)n * BB + b) * (size_t)DD;

        float x[7], tt[7];
        float s = 0.0f, ss = 0.0f;
        #pragma unroll
        for (int i = 0; i < 7; ++i) {
            const int d = tid + 256 * i;
            x[i] = xrow[d];
            const float t0 = x[i] + s_ptok[d];
            tt[i] = t0; s += t0; ss += t0 * t0;
        }
        if (n + 1 < NN) {       // hide HBM latency of the next row behind reductions
            #pragma unroll
            for (int i = 0; i < 7; ++i)
                __builtin_prefetch(xrow + (size_t)BB * DD + tid + 256 * i, 0, 0);
        }

        // wave32 butterfly + cross-wave reduction for mean / variance
        #pragma unroll
        for (int off = 16; off > 0; off >>= 1) {
            s  += __shfl_xor(s,  off, 32);
            ss += __shfl_xor(ss, off, 32);
        }
        if (lane == 0) { s_red[0][wv] = s; s_red[1][wv] = ss; }
        __syncthreads();
        float tot = 0.0f, tot2 = 0.0f;
        #pragma unroll
        for (int ww = 0; ww < 8; ++ww) { tot += s_red[0][ww]; tot2 += s_red[1][ww]; }
        const float mu   = tot * invD;
        const float var  = tot2 * invD - mu * mu;
        const float rstd = rsqrtf(var + 1e-5f);

        // normalize + partial dots against register-resident W_net slice
        float dp[PP];
        #pragma unroll
        for (int p = 0; p < PP; ++p) dp[p] = 0.0f;
        #pragma unroll
        for (int i = 0; i < 7; ++i) {
            const float v = (tt[i] - mu) * rstd * g[i] + be[i];
            #pragma unroll
            for (int p = 0; p < PP; ++p) dp[p] += v * wn[p][i];
        }
        #pragma unroll
        for (int p = 0; p < PP; ++p)
            #pragma unroll
            for (int off = 16; off > 0; off >>= 1)
                dp[p] += __shfl_xor(dp[p], off, 32);
        if (lane == 0)
            #pragma unroll
            for (int p = 0; p < PP; ++p) s_dot[wv][p] = dp[p];
        __syncthreads();

        if (tid < PP) {
            float t = 0.0f;
            #pragma unroll
            for (int ww = 0; ww < 8; ++ww) t += s_dot[ww][tid];
            t += s_bnet[tid];
            s_w[tid] = 1.0f / (1.0f + __expf(-t));
        }
        __syncthreads();

        float wl[PP];
        #pragma unroll
        for (int p = 0; p < PP; ++p) wl[p] = s_w[p];
        #pragma unroll
        for (int i = 0; i < 7; ++i) {
            const int d = tid + 256 * i;
            float acc = x[i];
            #pragma unroll
            for (int p = 0; p < PP; ++p) acc += wl[p] * s_bank[p * DD + d];
            orow[d] = acc;
        }
    }
}

extern "C" void kernel_launch(void* const* d_in, const int* in_sizes, int n_in,
                              void* d_out, int out_size, void* d_ws, size_t ws_size,
                              hipStream_t stream) {
    const float* X       = (const float*)d_in[0];
    const float* task_id = (const float*)d_in[1];
    const float* W_tok   = (const float*)d_in[2];
    const float* b_tok   = (const float*)d_in[3];
    const float* W_ptok  = (const float*)d_in[4];
    const float* b_ptok  = (const float*)d_in[5];
    const float* gamma   = (const float*)d_in[6];
    const float* beta    = (const float*)d_in[7];
    const float* W_net   = (const float*)d_in[8];
    const float* b_net   = (const float*)d_in[9];
    float* out = (float*)d_out;

    float* ws_bank = (float*)d_ws;                    // BB*PD floats  (~18.3 MB)
    float* ws_ptok = ws_bank + (size_t)BB * PD;       // BB*DD floats  (~1.8 MB)

    const int tiles   = (COLS / 16) * (BB / 16);      // 19712 (exact)
    const int blocksA = tiles / 8;                    // 8 waves per 256-thread block

    gpf_bank_wmma<<<blocksA, 256, 0, stream>>>(task_id, W_tok, b_tok,
                                               W_ptok, b_ptok, ws_bank, ws_ptok);
    gpf_fused_main<<<BB, 256, 0, stream>>>(X, gamma, beta, W_net, b_net,
                                           ws_bank, ws_ptok, out);
}